// GCNNet_2370821947637
// MI455X (gfx1250) — compile-verified
//
#include <hip/hip_runtime.h>

#define N_NODES   100000
#define N_EDGES   1000000
#define N_GRAPHS  2000
#define NODE_IN   163
#define HIDDEN    64
#define NEG_SLOPE 0.01f

typedef __attribute__((ext_vector_type(2))) float v2f;
typedef __attribute__((ext_vector_type(8))) float v8f;

__device__ __forceinline__ float lrelu(float x) {
    return x > 0.0f ? x : NEG_SLOPE * x;
}

// Native CDNA5 fp32 atomic add (device scope, non-returning) — avoids any CAS expansion.
__device__ __forceinline__ void gatomic_add_f32(float* p, float v) {
    asm volatile("global_atomic_add_f32 %0, %1, off scope:SCOPE_DEV"
                 :: "v"(p), "v"(v) : "memory");
}

// ---------------- degree / normalization ----------------
__global__ void k_deg_init(float* deg) {
    int i = blockIdx.x * blockDim.x + threadIdx.x;
    if (i < N_NODES) deg[i] = 1.0f;   // self-loop contributes 1
}

__global__ void k_deg_count(const int* __restrict__ dst, float* deg) {
    int e = blockIdx.x * blockDim.x + threadIdx.x;
    if (e < N_EDGES) gatomic_add_f32(&deg[dst[e]], 1.0f);
}

__global__ void k_rsqrt_inplace(float* deg) {
    int i = blockIdx.x * blockDim.x + threadIdx.x;
    if (i < N_NODES) deg[i] = rsqrtf(deg[i]);
}

// ---------------- WMMA f32 GEMM: out[N_NODES,64] = A[N_NODES,K] @ W[K,64] ----------------
// One wave per 16-row tile; 4 N-tiles of 16 cols; K-loop of V_WMMA_F32_16X16X4_F32.
// A frag (16x4 f32, ISA 7.12.2): lane<16 -> {K=kk,kk+1}, lane>=16 -> {K=kk+2,kk+3}, M=lane&15.
// B frag (4x16 f32): symmetric, N=lane&15.
// C/D (16x16 f32): VGPR r holds M=r (lanes 0-15) / M=r+8 (lanes 16-31), N=lane&15.
// Main loop is branch-free; the K%4 tail uses clamped indices + value masking (no EXEC divergence).
template<int K>
__global__ void k_gemm_wmma(const float* __restrict__ A, const float* __restrict__ W,
                            float* __restrict__ out) {
    const int wave = threadIdx.x >> 5;
    const int lane = threadIdx.x & 31;
    const int row0 = (blockIdx.x * 4 + wave) * 16;
    if (row0 >= N_NODES) return;

    const int m     = row0 + (lane & 15);
    const int khalf = (lane >> 4) << 1;       // 0 for lanes 0-15, 2 for lanes 16-31
    const int nlo   = lane & 15;
    const float* __restrict__ Arow = A + (size_t)m * K;

    v8f acc0 = {}, acc1 = {}, acc2 = {}, acc3 = {};

    constexpr int K4 = K & ~3;
    for (int kk = 0; kk < K4; kk += 4) {
        const int ka = kk + khalf;
        v2f a;
        if constexpr ((K & 1) == 0) {
            a = *(const v2f*)(Arow + ka);          // 8B-aligned -> global_load_b64
        } else {
            a.x = Arow[ka];
            a.y = Arow[ka + 1];
        }
        const float* __restrict__ Wr0 = W + ka * HIDDEN;
        const float* __restrict__ Wr1 = Wr0 + HIDDEN;
        v2f b0, b1, b2, b3;
        b0.x = Wr0[nlo];      b0.y = Wr1[nlo];
        b1.x = Wr0[nlo + 16]; b1.y = Wr1[nlo + 16];
        b2.x = Wr0[nlo + 32]; b2.y = Wr1[nlo + 32];
        b3.x = Wr0[nlo + 48]; b3.y = Wr1[nlo + 48];
        acc0 = __builtin_amdgcn_wmma_f32_16x16x4_f32(false, a, false, b0, (short)0, acc0, false, false);
        acc1 = __builtin_amdgcn_wmma_f32_16x16x4_f32(false, a, false, b1, (short)0, acc1, false, false);
        acc2 = __builtin_amdgcn_wmma_f32_16x16x4_f32(false, a, false, b2, (short)0, acc2, false, false);
        acc3 = __builtin_amdgcn_wmma_f32_16x16x4_f32(false, a, false, b3, (short)0, acc3, false, false);
    }

    if constexpr (K4 < K) {
        const int   ka0r = K4 + khalf;
        const int   ka1r = ka0r + 1;
        const int   ka0  = (ka0r < K) ? ka0r : (K - 1);   // clamped, always in-bounds
        const int   ka1  = (ka1r < K) ? ka1r : (K - 1);
        const float m0   = (ka0r < K) ? 1.0f : 0.0f;      // value mask (v_cndmask, no branch)
        const float m1   = (ka1r < K) ? 1.0f : 0.0f;
        v2f a;
        a.x = Arow[ka0];
        a.y = Arow[ka1];
        const float* __restrict__ Wr0 = W + ka0 * HIDDEN;
        const float* __restrict__ Wr1 = W + ka1 * HIDDEN;
        v2f b0, b1, b2, b3;
        b0.x = Wr0[nlo]      * m0;  b0.y = Wr1[nlo]      * m1;
        b1.x = Wr0[nlo + 16] * m0;  b1.y = Wr1[nlo + 16] * m1;
        b2.x = Wr0[nlo + 32] * m0;  b2.y = Wr1[nlo + 32] * m1;
        b3.x = Wr0[nlo + 48] * m0;  b3.y = Wr1[nlo + 48] * m1;
        acc0 = __builtin_amdgcn_wmma_f32_16x16x4_f32(false, a, false, b0, (short)0, acc0, false, false);
        acc1 = __builtin_amdgcn_wmma_f32_16x16x4_f32(false, a, false, b1, (short)0, acc1, false, false);
        acc2 = __builtin_amdgcn_wmma_f32_16x16x4_f32(false, a, false, b2, (short)0, acc2, false, false);
        acc3 = __builtin_amdgcn_wmma_f32_16x16x4_f32(false, a, false, b3, (short)0, acc3, false, false);
    }

    const int mrow = row0 + 8 * (lane >> 4);
#pragma unroll
    for (int r = 0; r < 8; ++r) {
        float* o = out + (size_t)(mrow + r) * HIDDEN + nlo;
        o[0]  = acc0[r];
        o[16] = acc1[r];
        o[32] = acc2[r];
        o[48] = acc3[r];
    }
}

// ---------------- aggregation ----------------
// agg[i,f] = d_i^-1 * hW[i,f] + b[f]   (self-loop term + bias)
__global__ void k_agg_init(const float* __restrict__ hW, const float* __restrict__ dis,
                           const float* __restrict__ bias, float* __restrict__ agg) {
    int idx = blockIdx.x * blockDim.x + threadIdx.x;
    if (idx < N_NODES * HIDDEN) {
        int i = idx >> 6, f = idx & 63;
        float d = dis[i];
        agg[idx] = d * d * hW[idx] + bias[f];
    }
}

// one wave per edge, lane handles 2 features; atomics resolve in L2 (25.6MB buffer << 192MB L2)
__global__ void k_edge_agg(const int* __restrict__ src, const int* __restrict__ dst,
                           const float* __restrict__ hW, const float* __restrict__ dis,
                           float* __restrict__ agg) {
    int gid = blockIdx.x * blockDim.x + threadIdx.x;   // [0, N_EDGES*32)
    int e = gid >> 5;
    if (e >= N_EDGES) return;
    int f = (gid & 31) * 2;
    int s = src[e], d = dst[e];
    float c = dis[s] * dis[d];
    float2 hv = *(const float2*)(hW + (size_t)s * HIDDEN + f);
    float* ap = agg + (size_t)d * HIDDEN + f;
    gatomic_add_f32(ap,     c * hv.x);
    gatomic_add_f32(ap + 1, c * hv.y);
}

__global__ void k_lrelu_inplace(float* a, int n) {
    int i = blockIdx.x * blockDim.x + threadIdx.x;
    if (i < n) a[i] = lrelu(a[i]);
}

// ---------------- pooling ----------------
__global__ void k_zero(float* p, int n) {
    int i = blockIdx.x * blockDim.x + threadIdx.x;
    if (i < n) p[i] = 0.0f;
}

__global__ void k_pool_sum(const float* __restrict__ h, const int* __restrict__ batch,
                           float* __restrict__ sums) {
    int idx = blockIdx.x * blockDim.x + threadIdx.x;
    if (idx < N_NODES * HIDDEN) {
        int i = idx >> 6, f = idx & 63;
        gatomic_add_f32(&sums[batch[i] * HIDDEN + f], h[idx]);
    }
}

__global__ void k_pool_cnt(const int* __restrict__ batch, float* __restrict__ cnts) {
    int i = blockIdx.x * blockDim.x + threadIdx.x;
    if (i < N_NODES) gatomic_add_f32(&cnts[batch[i]], 1.0f);
}

// ---------------- MLP head: one 64-thread block per graph ----------------
__global__ void k_head(const float* __restrict__ sums, const float* __restrict__ cnts,
                       const float* __restrict__ fc1_w, const float* __restrict__ fc1_b,
                       const float* __restrict__ fc2_w, const float* __restrict__ fc2_b,
                       float* __restrict__ out) {
    __shared__ float p[HIDDEN];
    __shared__ float red[HIDDEN];
    int g = blockIdx.x, t = threadIdx.x;
    float c = fmaxf(cnts[g], 1.0f);
    p[t] = sums[g * HIDDEN + t] / c;
    __syncthreads();
    float acc = fc1_b[t];
#pragma unroll 8
    for (int j = 0; j < HIDDEN; ++j) acc += p[j] * fc1_w[j * HIDDEN + t];
    red[t] = lrelu(acc) * fc2_w[t];
    __syncthreads();
    for (int s = 32; s > 0; s >>= 1) {
        if (t < s) red[t] += red[t + s];
        __syncthreads();
    }
    if (t == 0) out[g] = red[0] + fc2_b[0];
}

// ---------------- host ----------------
extern "C" void kernel_launch(void* const* d_in, const int* in_sizes, int n_in,
                              void* d_out, int out_size, void* d_ws, size_t ws_size,
                              hipStream_t stream) {
    (void)in_sizes; (void)n_in; (void)out_size; (void)ws_size;

    const float* x     = (const float*)d_in[0];
    const int*   ei    = (const int*)d_in[1];
    const int*   src   = ei;
    const int*   dst   = ei + N_EDGES;
    const int*   batch = (const int*)d_in[2];
    const float* Ws[3] = { (const float*)d_in[3], (const float*)d_in[5], (const float*)d_in[7] };
    const float* bs[3] = { (const float*)d_in[4], (const float*)d_in[6], (const float*)d_in[8] };
    const float* fc1_w = (const float*)d_in[9];
    const float* fc1_b = (const float*)d_in[10];
    const float* fc2_w = (const float*)d_in[11];
    const float* fc2_b = (const float*)d_in[12];
    float* out = (float*)d_out;

    // workspace carve-up (256B aligned regions)
    char* ws = (char*)d_ws;
    size_t off = 0;
    auto carve = [&](size_t bytes) -> float* {
        float* p = (float*)(ws + off);
        off = (off + bytes + 255) & ~(size_t)255;
        return p;
    };
    float* dis  = carve((size_t)N_NODES * sizeof(float));
    float* hW   = carve((size_t)N_NODES * HIDDEN * sizeof(float));
    float* aggA = carve((size_t)N_NODES * HIDDEN * sizeof(float));
    float* aggB = carve((size_t)N_NODES * HIDDEN * sizeof(float));
    float* sums = carve((size_t)N_GRAPHS * HIDDEN * sizeof(float));
    float* cnts = carve((size_t)N_GRAPHS * sizeof(float));

    const int NB = 256;
    const int NH = N_NODES * HIDDEN;

    // degrees -> dis = rsqrt(1 + indeg)
    k_deg_init<<<(N_NODES + NB - 1) / NB, NB, 0, stream>>>(dis);
    k_deg_count<<<(N_EDGES + NB - 1) / NB, NB, 0, stream>>>(dst, dis);
    k_rsqrt_inplace<<<(N_NODES + NB - 1) / NB, NB, 0, stream>>>(dis);

    // 3 GCN layers
    const float* hin = x;
    float* aggs[3] = { aggA, aggB, aggA };
    const int mtiles = (N_NODES + 15) / 16;            // 6250
    const int gemm_blocks = (mtiles + 3) / 4;          // 4 waves/block
    for (int L = 0; L < 3; ++L) {
        float* agg = aggs[L];
        if (L == 0)
            k_gemm_wmma<NODE_IN><<<gemm_blocks, 128, 0, stream>>>(hin, Ws[L], hW);
        else
            k_gemm_wmma<HIDDEN><<<gemm_blocks, 128, 0, stream>>>(hin, Ws[L], hW);
        k_agg_init<<<(NH + NB - 1) / NB, NB, 0, stream>>>(hW, dis, bs[L], agg);
        k_edge_agg<<<(N_EDGES * 32) / NB, NB, 0, stream>>>(src, dst, hW, dis, agg);
        k_lrelu_inplace<<<(NH + NB - 1) / NB, NB, 0, stream>>>(agg, NH);
        hin = agg;
    }

    // global mean pool + head
    k_zero<<<(N_GRAPHS * HIDDEN + NB - 1) / NB, NB, 0, stream>>>(sums, N_GRAPHS * HIDDEN);
    k_zero<<<(N_GRAPHS + NB - 1) / NB, NB, 0, stream>>>(cnts, N_GRAPHS);
    k_pool_sum<<<(NH + NB - 1) / NB, NB, 0, stream>>>(hin, batch, sums);
    k_pool_cnt<<<(N_NODES + NB - 1) / NB, NB, 0, stream>>>(batch, cnts);
    k_head<<<N_GRAPHS, HIDDEN, 0, stream>>>(sums, cnts, fc1_w, fc1_b, fc2_w, fc2_b, out);
}